// MachampCRFDecoder_19464791786114
// MI455X (gfx1250) — compile-verified
//
#include <hip/hip_runtime.h>
#include <hip/hip_bf16.h>

typedef __attribute__((ext_vector_type(2))) float v2f;
typedef __attribute__((ext_vector_type(8))) float v8f;

#define BB 64
#define TT 512
#define DD 1024
#define KK 17
#define NEGF (-10000.0f)

// ---------------------------------------------------------------------------
// Kernel 0: pack W (D x 17, row-major) into B-fragment layout for
// V_WMMA_F32_16X16X4_F32.  B is 4x16 (KxN) per fragment:
//   lanes 0-15 : v0 = B[k+0][n], v1 = B[k+1][n]   (n = lane)
//   lanes 16-31: v0 = B[k+2][n], v1 = B[k+3][n]   (n = lane-16)
// Layout: Bpack[((k4*2 + ntile)*32 + lane)*2 + v], 256 k-steps, 2 N-tiles.
// ---------------------------------------------------------------------------
__global__ void pack_W_kernel(const float* __restrict__ W, float* __restrict__ Bpack) {
    int idx = blockIdx.x * blockDim.x + threadIdx.x;   // 16384 total
    if (idx >= 256 * 2 * 32) return;
    int lane = idx & 31;
    int nt   = (idx >> 5) & 1;
    int k4   = idx >> 6;
    int n    = nt * 16 + (lane & 15);
    int kb   = k4 * 4 + (lane >> 4) * 2;
    float b0 = 0.f, b1 = 0.f;
    if (n < KK) {
        b0 = W[(size_t)(kb + 0) * KK + n];
        b1 = W[(size_t)(kb + 1) * KK + n];
    }
    Bpack[(size_t)idx * 2 + 0] = b0;
    Bpack[(size_t)idx * 2 + 1] = b1;
}

// ---------------------------------------------------------------------------
// Kernel 1: logits = mlm_out @ W + b   via V_WMMA_F32_16X16X4_F32.
// One wave per 16-row M-tile; each wave computes a 16x32 tile (2 WMMA chains),
// only columns 0..16 are stored.
// ---------------------------------------------------------------------------
__global__ __launch_bounds__(256) void gemm_logits_kernel(
    const float* __restrict__ A,      // (32768, 1024)
    const float* __restrict__ Bpack,  // packed B fragments
    const float* __restrict__ bias,   // (17,)
    float* __restrict__ logits)       // (32768, 17)
{
    int wave = (int)((blockIdx.x * blockDim.x + threadIdx.x) >> 5);
    int lane = threadIdx.x & 31;
    size_t row_base = (size_t)wave * 16;

    // A fragment source: lane l<16 -> row l, cols k..k+1 ; l>=16 -> row l-16, cols k+2..k+3
    const float* Arow = A + (row_base + (lane & 15)) * DD + (lane >> 4) * 2;

    v8f acc0 = {};   // N-tile 0 (cols 0..15)
    v8f acc1 = {};   // N-tile 1 (cols 16..31, only 16 kept)

#pragma unroll 4
    for (int k4 = 0; k4 < 256; ++k4) {
        v2f a  = *(const v2f*)(Arow + k4 * 4);
        v2f b0 = *(const v2f*)(Bpack + ((size_t)(k4 * 2 + 0) * 32 + lane) * 2);
        v2f b1 = *(const v2f*)(Bpack + ((size_t)(k4 * 2 + 1) * 32 + lane) * 2);
        __builtin_prefetch(Arow + k4 * 4 + 128, 0, 0);
        acc0 = __builtin_amdgcn_wmma_f32_16x16x4_f32(false, a, false, b0,
                                                     (short)0, acc0, false, false);
        acc1 = __builtin_amdgcn_wmma_f32_16x16x4_f32(false, a, false, b1,
                                                     (short)0, acc1, false, false);
    }

    // C/D layout: VGPR i, lanes 0-15 -> M=i, lanes 16-31 -> M=8+i; N = lane&15 (+16*nt)
    int n0    = lane & 15;
    int mtop  = (lane >> 4) * 8;
    float bi0 = bias[n0];
    bool  w1  = (n0 == 0);            // N-tile1: only column 16 is real
    float bi1 = w1 ? bias[16] : 0.f;

#pragma unroll
    for (int i = 0; i < 8; ++i) {
        size_t row = row_base + mtop + i;
        logits[row * KK + n0] = acc0[i] + bi0;
        if (w1) logits[row * KK + 16] = acc1[i] + bi1;
    }
}

// ---------------------------------------------------------------------------
// Kernel 2: per-batch CRF — gold score, forward logZ, Viterbi + backtrace.
// One wave (32 threads) per batch element; lane = tag (17 valid).
// ---------------------------------------------------------------------------
__global__ __launch_bounds__(32) void crf_kernel(
    const float* __restrict__ logits,
    const int*   __restrict__ mask,
    const int*   __restrict__ gold,
    const float* __restrict__ trans,
    const float* __restrict__ start_trans,
    const float* __restrict__ end_trans,
    const float* __restrict__ t_allowed,
    const float* __restrict__ s_allowed,
    const float* __restrict__ e_allowed,
    unsigned char* __restrict__ bp,     // [B][T-1][32]
    float* __restrict__ partial,        // [B] : score - logZ
    float* __restrict__ pred_out)       // [B*T] (as float)
{
    int b    = blockIdx.x;
    int lane = threadIdx.x;
    int j    = lane;
    bool valid = (j < KK);
    int  jj  = valid ? j : 0;

    __shared__ float sbuf[32];

    const int*   maskb = mask + b * TT;
    const int*   goldb = gold + b * TT;
    const float* logb  = logits + (size_t)b * TT * KK;

    // ---- gold path score (lane-parallel over t) ----
    float sc = 0.f;
    float lenp = 0.f;
    for (int t = lane; t < TT; t += 32) {
        int m = maskb[t];
        if (m) {
            lenp += 1.f;
            int g = goldb[t];
            if (t < TT - 1) sc += logb[(size_t)t * KK + g];
            if (t >= 1)     sc += trans[goldb[t - 1] * KK + g];
        }
    }
    sbuf[lane] = sc; __syncthreads();
    for (int off = 16; off > 0; off >>= 1) {
        if (lane < off) sbuf[lane] += sbuf[lane + off];
        __syncthreads();
    }
    float scsum = sbuf[0]; __syncthreads();
    sbuf[lane] = lenp; __syncthreads();
    for (int off = 16; off > 0; off >>= 1) {
        if (lane < off) sbuf[lane] += sbuf[lane + off];
        __syncthreads();
    }
    int len = (int)(sbuf[0] + 0.5f); __syncthreads();

    float score = 0.f;
    if (lane == 0) {
        int lt = goldb[len - 1];
        score = scsum + start_trans[goldb[0]] + end_trans[lt];
        if (maskb[TT - 1]) score += logb[(size_t)(TT - 1) * KK + lt];
    }

    // per-lane transition column (i -> j) and constrained version
    float tcol[KK], ctcol[KK];
#pragma unroll
    for (int i = 0; i < KK; ++i) {
        float tr = trans[i * KK + jj];
        float al = t_allowed[i * KK + jj];
        tcol[i]  = tr;
        ctcol[i] = tr * al + NEGF * (1.f - al);
    }

    // ---- forward algorithm ----
    float alpha = valid ? (start_trans[j] + logb[j]) : NEGF;
    for (int t = 1; t < TT; ++t) {
        sbuf[lane] = alpha; __syncthreads();
        float av[KK];
#pragma unroll
        for (int i = 0; i < KK; ++i) av[i] = sbuf[i] + tcol[i];
        float m = av[0];
#pragma unroll
        for (int i = 1; i < KK; ++i) m = fmaxf(m, av[i]);
        float s = 0.f;
#pragma unroll
        for (int i = 0; i < KK; ++i) s += __expf(av[i] - m);
        float na = m + __logf(s) + logb[(size_t)t * KK + jj];
        if (maskb[t] && valid) alpha = na;
        __syncthreads();
    }
    sbuf[lane] = valid ? (alpha + end_trans[jj]) : NEGF; __syncthreads();
    if (lane == 0) {
        float m = sbuf[0];
        for (int i = 1; i < KK; ++i) m = fmaxf(m, sbuf[i]);
        float s = 0.f;
        for (int i = 0; i < KK; ++i) s += __expf(sbuf[i] - m);
        float logZ = m + __logf(s);
        partial[b] = score - logZ;
    }
    __syncthreads();

    // ---- Viterbi ----
    float sal = valid ? s_allowed[jj] : 0.f;
    float v = valid ? (start_trans[jj] * sal + NEGF * (1.f - sal) + logb[jj]) : NEGF;
    for (int t = 1; t < TT; ++t) {
        sbuf[lane] = v; __syncthreads();
        float best = sbuf[0] + ctcol[0];
        int bpi = 0;
#pragma unroll
        for (int i = 1; i < KK; ++i) {
            float val = sbuf[i] + ctcol[i];
            if (val > best) { best = val; bpi = i; }   // first-max semantics
        }
        if (valid)
            bp[((size_t)b * (TT - 1) + (t - 1)) * 32 + j] = (unsigned char)bpi;
        float nv = best + logb[(size_t)t * KK + jj];
        if (maskb[t] && valid) v = nv;
        __syncthreads();
    }
    sbuf[lane] = v; __syncthreads();
    if (lane == 0) {
        int tag = 0;
        float bestv = -3.4e38f;
        for (int q = 0; q < KK; ++q) {
            float eal = e_allowed[q];
            float ce  = end_trans[q] * eal + NEGF * (1.f - eal);
            float val = sbuf[q] + ce;
            if (val > bestv) { bestv = val; tag = q; }
        }
        // backtrace: pred[t] = tag_t * mask[t]
        for (int t = TT - 1; t >= 1; --t) {
            pred_out[b * TT + t] = maskb[t] ? (float)tag : 0.f;
            int prev = maskb[t] ? (int)bp[((size_t)b * (TT - 1) + (t - 1)) * 32 + tag] : tag;
            tag = prev;
        }
        pred_out[b * TT + 0] = maskb[0] ? (float)tag : 0.f;
    }
}

// ---------------------------------------------------------------------------
// Kernel 3: loss = -sum_b(score_b - logZ_b)
// ---------------------------------------------------------------------------
__global__ __launch_bounds__(64) void loss_reduce_kernel(const float* __restrict__ partial,
                                                         float* __restrict__ loss) {
    __shared__ float s[64];
    s[threadIdx.x] = partial[threadIdx.x];
    __syncthreads();
    for (int off = 32; off > 0; off >>= 1) {
        if (threadIdx.x < off) s[threadIdx.x] += s[threadIdx.x + off];
        __syncthreads();
    }
    if (threadIdx.x == 0) loss[0] = -s[0];
}

// ---------------------------------------------------------------------------
extern "C" void kernel_launch(void* const* d_in, const int* in_sizes, int n_in,
                              void* d_out, int out_size, void* d_ws, size_t ws_size,
                              hipStream_t stream) {
    const float* mlm   = (const float*)d_in[0];
    const int*   mask  = (const int*)  d_in[1];
    const int*   gold  = (const int*)  d_in[2];
    const float* W     = (const float*)d_in[3];
    const float* bias  = (const float*)d_in[4];
    const float* trans = (const float*)d_in[5];
    const float* st    = (const float*)d_in[6];
    const float* et    = (const float*)d_in[7];
    const float* ta    = (const float*)d_in[8];
    const float* sa    = (const float*)d_in[9];
    const float* ea    = (const float*)d_in[10];

    float* out    = (float*)d_out;
    float* logits = out;                                    // 64*512*17
    float* loss   = out + (size_t)BB * TT * KK;             // 1
    float* pred   = loss + 1;                               // 64*512

    float*         Bpack   = (float*)d_ws;                                  // 128 KB
    float*         partial = (float*)((char*)d_ws + 131072);                // 256 B
    unsigned char* bpbuf   = (unsigned char*)d_ws + 131072 + 256;           // ~1 MB

    pack_W_kernel<<<64, 256, 0, stream>>>(W, Bpack);
    gemm_logits_kernel<<<256, 256, 0, stream>>>(mlm, Bpack, bias, logits);
    crf_kernel<<<BB, 32, 0, stream>>>(logits, mask, gold, trans, st, et,
                                      ta, sa, ea, bpbuf, partial, pred);
    loss_reduce_kernel<<<1, 64, 0, stream>>>(partial, loss);
}